// MeshVertexNormals_66511863546459
// MI455X (gfx1250) — compile-verified
//
#include <hip/hip_runtime.h>
#include <hip/hip_bf16.h>

// Mesh vertex normals, B=16, V=500000, F=1000000.
// d_in[0]: vertices  (B*V*3 f32)
// d_in[1]: faces     (F*3 int)
// d_out  : [vectors B*V*3 f32][areas B*F f32]
//
// Strategy (MI455X): vertices (96MB) + normal accumulator (96MB) both live in
// the 192MB L2; faces are streamed with non-temporal loads and areas with
// non-temporal stores so they don't evict the hot set. Scatter-add uses the
// native global_atomic_add_f32 (resolves in L2). ~0.5GB HBM traffic total
// => ~25us floor at 23.3 TB/s.

typedef float v4f __attribute__((ext_vector_type(4)));

namespace {
constexpr int       kB   = 16;
constexpr int       kV   = 500000;
constexpr int       kF   = 1000000;
constexpr float     kEps = 1e-6f;
constexpr long long kVecElems = (long long)kB * kV * 3; // 24,000,000
}

__global__ __launch_bounds__(256) void zero_f4_kernel(v4f* __restrict__ p, int n4) {
    int i = blockIdx.x * blockDim.x + threadIdx.x;
    if (i < n4) p[i] = (v4f){0.f, 0.f, 0.f, 0.f};
}

__global__ __launch_bounds__(256) void face_scatter_kernel(
    const float* __restrict__ verts,   // B*V*3
    const int*   __restrict__ faces,   // F*3 (int32)
    float*       __restrict__ vacc,    // B*V*3, pre-zeroed accumulator (== output vectors)
    float*       __restrict__ areas)   // B*F
{
    int f = blockIdx.x * blockDim.x + threadIdx.x;
    if (f >= kF) return;

    // Stream the faces array: read-once -> non-temporal; prefetch ahead
    // (gfx1250 global_prefetch_b8) to cover the strided int3 pattern.
    __builtin_prefetch(faces + 3 * (size_t)f + 256, 0, 0);
    int i0 = __builtin_nontemporal_load(faces + 3 * (size_t)f + 0);
    int i1 = __builtin_nontemporal_load(faces + 3 * (size_t)f + 1);
    int i2 = __builtin_nontemporal_load(faces + 3 * (size_t)f + 2);

    const size_t o0 = (size_t)i0 * 3;
    const size_t o1 = (size_t)i1 * 3;
    const size_t o2 = (size_t)i2 * 3;

#pragma unroll 4
    for (int b = 0; b < kB; ++b) {
        const float* __restrict__ vb = verts + (size_t)b * (size_t)kV * 3;
        // Gather the three vertices (hits L2: 96MB vertex array is resident).
        float ax = vb[o0 + 0], ay = vb[o0 + 1], az = vb[o0 + 2];
        float bx = vb[o1 + 0], by = vb[o1 + 1], bz = vb[o1 + 2];
        float cx = vb[o2 + 0], cy = vb[o2 + 1], cz = vb[o2 + 2];

        // n = cross(v1 - v0, v2 - v1)
        float e0x = bx - ax, e0y = by - ay, e0z = bz - az;
        float e1x = cx - bx, e1y = cy - by, e1z = cz - bz;
        float nx = e0y * e1z - e0z * e1y;
        float ny = e0z * e1x - e0x * e1z;
        float nz = e0x * e1y - e0y * e1x;

        // area = 0.5 * |n|  (write-once stream -> non-temporal store)
        float area = 0.5f * sqrtf(nx * nx + ny * ny + nz * nz);
        __builtin_nontemporal_store(area, areas + (size_t)b * kF + f);

        // Scatter-add into the L2-resident accumulator via native f32 atomics.
        float* __restrict__ ab = vacc + (size_t)b * (size_t)kV * 3;
        unsafeAtomicAdd(ab + o0 + 0, nx);
        unsafeAtomicAdd(ab + o0 + 1, ny);
        unsafeAtomicAdd(ab + o0 + 2, nz);
        unsafeAtomicAdd(ab + o1 + 0, nx);
        unsafeAtomicAdd(ab + o1 + 1, ny);
        unsafeAtomicAdd(ab + o1 + 2, nz);
        unsafeAtomicAdd(ab + o2 + 0, nx);
        unsafeAtomicAdd(ab + o2 + 1, ny);
        unsafeAtomicAdd(ab + o2 + 2, nz);
    }
}

// Normalize 4 vertices (12 floats) per thread with 3x b128 loads/stores.
// nvert (8M) is divisible by 4, so no tail handling needed.
__global__ __launch_bounds__(256) void normalize4_kernel(float* __restrict__ v,
                                                         int ngroup) {
    int g = blockIdx.x * blockDim.x + threadIdx.x;
    if (g >= ngroup) return;

    v4f* p = (v4f*)(v + (size_t)g * 12);
    v4f a = p[0];   // v0.xyz v1.x
    v4f b = p[1];   // v1.yz  v2.xy
    v4f c = p[2];   // v2.z   v3.xyz

    // vertex 0: a.x a.y a.z
    float s0 = 1.0f / fmaxf(sqrtf(a.x * a.x + a.y * a.y + a.z * a.z), kEps);
    // vertex 1: a.w b.x b.y
    float s1 = 1.0f / fmaxf(sqrtf(a.w * a.w + b.x * b.x + b.y * b.y), kEps);
    // vertex 2: b.z b.w c.x
    float s2 = 1.0f / fmaxf(sqrtf(b.z * b.z + b.w * b.w + c.x * c.x), kEps);
    // vertex 3: c.y c.z c.w
    float s3 = 1.0f / fmaxf(sqrtf(c.y * c.y + c.z * c.z + c.w * c.w), kEps);

    a.x *= s0; a.y *= s0; a.z *= s0;
    a.w *= s1; b.x *= s1; b.y *= s1;
    b.z *= s2; b.w *= s2; c.x *= s2;
    c.y *= s3; c.z *= s3; c.w *= s3;

    // Final output: write-once, never re-read -> non-temporal b128 stores.
    __builtin_nontemporal_store(a, p + 0);
    __builtin_nontemporal_store(b, p + 1);
    __builtin_nontemporal_store(c, p + 2);
}

extern "C" void kernel_launch(void* const* d_in, const int* in_sizes, int n_in,
                              void* d_out, int out_size, void* d_ws, size_t ws_size,
                              hipStream_t stream) {
    (void)in_sizes; (void)n_in; (void)out_size; (void)d_ws; (void)ws_size;

    const float* verts = (const float*)d_in[0];
    const int*   faces = (const int*)d_in[1];

    float* out   = (float*)d_out;
    float* vacc  = out;                              // vectors region (B*V*3)
    float* areas = out + (size_t)kVecElems;          // areas region (B*F)

    // 1) Zero the accumulator (d_out is poisoned between runs).
    int n4 = (int)(kVecElems / 4);                   // 6,000,000 float4s
    zero_f4_kernel<<<(n4 + 255) / 256, 256, 0, stream>>>((v4f*)vacc, n4);

    // 2) Per-face: cross product, area store, 9x atomic scatter-add x 16 batches.
    face_scatter_kernel<<<(kF + 255) / 256, 256, 0, stream>>>(verts, faces, vacc, areas);

    // 3) Normalize vertex normals in place, 4 vertices / thread.
    int ngroup = (int)((long long)kB * kV / 4);      // 2,000,000
    normalize4_kernel<<<(ngroup + 255) / 256, 256, 0, stream>>>(vacc, ngroup);
}